// MacroContextAdderToSubAST_41987600285769
// MI455X (gfx1250) — compile-verified
//
#include <hip/hip_runtime.h>

// f32 WMMA fragment types (wave32): A/B 16x4 & 4x16 f32 = 2 VGPRs, C/D 16x16 f32 = 8 VGPRs
typedef float v2f __attribute__((ext_vector_type(2)));
typedef float v8f __attribute__((ext_vector_type(8)));

#define D          256
#define TILE_ROWS  64    // 4 row-tiles of 16 -> each B fragment reused 4x, halves L2 W-traffic
#define LDS_STRIDE 260   // 260 % 64 == 4 -> consecutive rows shift banks, no 16-way conflicts

// ---------------------------------------------------------------------------
// Flag helpers: flags[row] = 1 iff row is scattered by the fused kernel.
// ---------------------------------------------------------------------------
__global__ __launch_bounds__(256) void zero_flags(unsigned int* __restrict__ flags, int n) {
    int i = blockIdx.x * blockDim.x + threadIdx.x;
    int stride = gridDim.x * blockDim.x;
    for (; i < n; i += stride) flags[i] = 0u;
}

__global__ __launch_bounds__(256) void set_flags(const int* __restrict__ keys,
                                                 unsigned int* __restrict__ flags, int M) {
    int i = blockIdx.x * blockDim.x + threadIdx.x;
    if (i < M) flags[keys[i]] = 1u;   // keys are unique; same-value stores are safe anyway
}

// Copy only rows NOT covered by the scatter (flag == 0). 4 rows per block,
// 64 threads per row (1 float4 each) -> flag is wave-uniform (no divergence).
__global__ __launch_bounds__(256) void copy_unflagged_rows(
    const float* __restrict__ prev, const unsigned int* __restrict__ flags,
    float* __restrict__ out, int n_rows) {
    int row = blockIdx.x * 4 + (threadIdx.x >> 6);
    if (row >= n_rows) return;
    if (flags[row] != 0u) return;
    int c4 = (threadIdx.x & 63) << 2;
    *(float4*)(out + (size_t)row * D + c4) =
        *(const float4*)(prev + (size_t)row * D + c4);
}

// ---------------------------------------------------------------------------
// Fused gather -> GEMM1(relu) -> GEMM2(sigmoid gate) -> blend -> scatter
// ---------------------------------------------------------------------------
__global__ __launch_bounds__(256) void fused_gated_update(
    const float* __restrict__ prev,   // [N_AST, 256]
    const float* __restrict__ cfg,    // [N_CFG, 256]
    const int*   __restrict__ keys,   // [M]
    const int*   __restrict__ vals,   // [M]
    const float* __restrict__ Wu,     // [256, 256]
    const float* __restrict__ bu,     // [256]
    const float* __restrict__ Wg,     // [512, 256]
    const float* __restrict__ bg,     // [256]
    float* __restrict__ out,          // [N_AST, 256]
    int M)
{
    extern __shared__ float smem[];
    float* lds_state = smem;                              // 64 x 260
    float* lds_upd   = smem + TILE_ROWS * LDS_STRIDE;     // 64 x 260
    float* lds_proj  = smem + 2 * TILE_ROWS * LDS_STRIDE; // 64 x 260
    __shared__ int s_keys[TILE_ROWS];

    const int tid  = threadIdx.x;
    const int m0   = blockIdx.x * TILE_ROWS;
    const int lane = tid & 31;
    const int wave = tid >> 5;
    const int half = lane >> 4;   // which 16-lane half
    const int ln   = lane & 15;

    if (tid < TILE_ROWS) {
        int m = m0 + tid;
        s_keys[tid] = (m < M) ? keys[m] : -1;
    }

    // ---- gather: 64 rows x 64 float4 per matrix; 256 threads x 16 iters ----
    for (int i = 0; i < 16; ++i) {
        int fidx = i * 256 + tid;
        int row  = fidx >> 6;          // /64 -> row in tile
        int c4   = (fidx & 63) << 2;   // col (floats)
        int m    = m0 + row;
        float4 sv = {0.f, 0.f, 0.f, 0.f};
        float4 uv = {0.f, 0.f, 0.f, 0.f};
        if (m < M) {
            int k = keys[m];
            int v = vals[m];
            sv = *(const float4*)(prev + (size_t)k * D + c4);
            uv = *(const float4*)(cfg  + (size_t)v * D + c4);
        }
        *(float4*)(lds_state + row * LDS_STRIDE + c4) = sv;
        *(float4*)(lds_upd   + row * LDS_STRIDE + c4) = uv;
    }
    __syncthreads();

    // ---- GEMM1: proj = relu(upd @ Wu + bu); each wave owns 2 N-tiles (32 cols) ----
    for (int nt = 0; nt < 2; ++nt) {
        const int col = wave * 32 + nt * 16 + ln;   // output column, B-matrix N = ln
        v8f acc[4];
        acc[0] = v8f{}; acc[1] = v8f{}; acc[2] = v8f{}; acc[3] = v8f{};
        #pragma unroll 4
        for (int kb = 0; kb < D; kb += 4) {
            const int ka = kb + 2 * half;           // this lane's K pair
            v2f b;
            b.x = Wu[(size_t)ka * D + col];
            b.y = Wu[(size_t)(ka + 1) * D + col];
            #pragma unroll
            for (int rt = 0; rt < 4; ++rt) {
                v2f a = *(const v2f*)(lds_upd + (rt * 16 + ln) * LDS_STRIDE + ka);
                acc[rt] = __builtin_amdgcn_wmma_f32_16x16x4_f32(false, a, false, b,
                                                                (short)0, acc[rt], false, false);
            }
        }
        const float biasv = bu[col];
        #pragma unroll
        for (int rt = 0; rt < 4; ++rt) {
            #pragma unroll
            for (int r = 0; r < 8; ++r) {
                const int mrow = rt * 16 + r + 8 * half;   // D-matrix row
                float p = acc[rt][r] + biasv;
                p = p > 0.f ? p : 0.f;
                lds_proj[mrow * LDS_STRIDE + col] = p;
            }
        }
    }
    __syncthreads();

    // ---- GEMM2: z = state @ Wg[0:256] + proj @ Wg[256:512] + bg; blend + scatter ----
    for (int nt = 0; nt < 2; ++nt) {
        const int col = wave * 32 + nt * 16 + ln;
        v8f acc[4];
        acc[0] = v8f{}; acc[1] = v8f{}; acc[2] = v8f{}; acc[3] = v8f{};
        #pragma unroll 4
        for (int kb = 0; kb < D; kb += 4) {         // K = 0..255 : state part
            const int ka = kb + 2 * half;
            v2f b;
            b.x = Wg[(size_t)ka * D + col];
            b.y = Wg[(size_t)(ka + 1) * D + col];
            #pragma unroll
            for (int rt = 0; rt < 4; ++rt) {
                v2f a = *(const v2f*)(lds_state + (rt * 16 + ln) * LDS_STRIDE + ka);
                acc[rt] = __builtin_amdgcn_wmma_f32_16x16x4_f32(false, a, false, b,
                                                                (short)0, acc[rt], false, false);
            }
        }
        #pragma unroll 4
        for (int kb = 0; kb < D; kb += 4) {         // K = 256..511 : proj part
            const int ka = kb + 2 * half;
            v2f b;
            b.x = Wg[(size_t)(D + ka) * D + col];
            b.y = Wg[(size_t)(D + ka + 1) * D + col];
            #pragma unroll
            for (int rt = 0; rt < 4; ++rt) {
                v2f a = *(const v2f*)(lds_proj + (rt * 16 + ln) * LDS_STRIDE + ka);
                acc[rt] = __builtin_amdgcn_wmma_f32_16x16x4_f32(false, a, false, b,
                                                                (short)0, acc[rt], false, false);
            }
        }
        const float bgv = bg[col];
        #pragma unroll
        for (int rt = 0; rt < 4; ++rt) {
            #pragma unroll
            for (int r = 0; r < 8; ++r) {
                const int mrow = rt * 16 + r + 8 * half;
                float z = acc[rt][r] + bgv;
                float g = 1.f / (1.f + __expf(-z));
                float s = lds_state[mrow * LDS_STRIDE + col];
                float p = lds_proj [mrow * LDS_STRIDE + col];
                float nv = fmaf(g, s - p, p);       // g*s + (1-g)*p
                int key = s_keys[mrow];
                if (key >= 0) out[(size_t)key * D + col] = nv;
            }
        }
    }
}

extern "C" void kernel_launch(void* const* d_in, const int* in_sizes, int n_in,
                              void* d_out, int out_size, void* d_ws, size_t ws_size,
                              hipStream_t stream) {
    const float* prev = (const float*)d_in[0];
    const float* cfg  = (const float*)d_in[1];
    const int*   keys = (const int*)d_in[2];
    const int*   vals = (const int*)d_in[3];
    const float* Wu   = (const float*)d_in[4];
    const float* bu   = (const float*)d_in[5];
    const float* Wg   = (const float*)d_in[6];
    const float* bg   = (const float*)d_in[7];
    float* out = (float*)d_out;

    const int M      = in_sizes[2];
    const int n_rows = in_sizes[0] / D;   // N_AST

    const size_t flag_bytes = (size_t)n_rows * sizeof(unsigned int);
    if (ws_size >= flag_bytes) {
        // Skip copying the 80% of rows that the scatter overwrites.
        unsigned int* flags = (unsigned int*)d_ws;
        hipLaunchKernelGGL(zero_flags, dim3(1024), dim3(256), 0, stream, flags, n_rows);
        hipLaunchKernelGGL(set_flags, dim3((M + 255) / 256), dim3(256), 0, stream,
                           keys, flags, M);
        hipLaunchKernelGGL(copy_unflagged_rows, dim3((n_rows + 3) / 4), dim3(256), 0, stream,
                           prev, flags, out, n_rows);
    } else {
        hipMemcpyAsync(out, prev, (size_t)in_sizes[0] * sizeof(float),
                       hipMemcpyDeviceToDevice, stream);
    }

    const int nblocks = (M + TILE_ROWS - 1) / TILE_ROWS;
    const size_t shmem = (size_t)3 * TILE_ROWS * LDS_STRIDE * sizeof(float);
    hipLaunchKernelGGL(fused_gated_update, dim3(nblocks), dim3(256), shmem, stream,
                       prev, cfg, keys, vals, Wu, bu, Wg, bg, out, M);
}